// GATLayer_90915867722295
// MI455X (gfx1250) — compile-verified
//
#include <hip/hip_runtime.h>
#include <hip/hip_bf16.h>

typedef float v2f __attribute__((ext_vector_type(2)));
typedef float v8f __attribute__((ext_vector_type(8)));

#define RREP   48      // B*S
#define NNODE  1000
#define FDIM   64
#define HHEADS 4
#define CDIM   64
#define OUTD   256     // H*C
#define NEDGE  12000
#define EPRIME 13000   // NEDGE + NNODE self loops
#define NROWS  48000   // RREP*NNODE
#define NEG_SLOPE 0.2f

// ---------------------------------------------------------------------------
// Kernel 1: out[r,n,f] = bias[f];  den[r,n,h] = 0
// ---------------------------------------------------------------------------
__global__ void gat_init(float* __restrict__ out, float* __restrict__ den,
                         const float* __restrict__ bias) {
    int tid = blockIdx.x * blockDim.x + threadIdx.x;
    if (tid < NROWS * FDIM) out[tid] = bias[tid & (FDIM - 1)];
    if (tid < NROWS * HHEADS) den[tid] = 0.0f;
}

// ---------------------------------------------------------------------------
// Kernel 2: h = x @ W  via V_WMMA_F32_16X16X4_F32
//   x : [48000, 64] row-major, W : [64, 256] row-major, h : [48000, 256]
//   grid.x = 3000 M-tiles; block = 128 threads = 4 waves; each wave owns a
//   64-wide N strip (4x 16x16 tiles) so the A fragment is reused 4x per kstep.
// ---------------------------------------------------------------------------
__global__ void gat_proj_wmma(const float* __restrict__ x,
                              const float* __restrict__ W,
                              float* __restrict__ h) {
    const int lane = threadIdx.x & 31;
    const int nblk = threadIdx.x >> 5;            // 0..3 -> cols nblk*64..+63
    const int mt   = blockIdx.x;                  // 0..2999
    const int mrow = mt * 16 + (lane & 15);       // A row for this lane
    const int kb   = (lane >> 4) * 2;             // K sub-offset: 0 or 2
    const int col0 = nblk * 64 + (lane & 15);     // B/D column for this lane

    v8f acc0 = {}, acc1 = {}, acc2 = {}, acc3 = {};

    #pragma unroll
    for (int kk = 0; kk < 16; ++kk) {
        const int k = kk * 4 + kb;                // even -> 8B aligned
        // A fragment 16x4 f32: VGPR0 = K=kb, VGPR1 = K=kb+1 (ISA 7.12.2)
        v2f a;
        const float2 av = *(const float2*)(x + (size_t)mrow * FDIM + k);
        a.x = av.x; a.y = av.y;
        // B fragment 4x16 f32: row K=kb / kb+1 striped across lanes 0-15/16-31
        v2f b0, b1, b2, b3;
        b0.x = W[k * OUTD + col0 +  0]; b0.y = W[(k + 1) * OUTD + col0 +  0];
        b1.x = W[k * OUTD + col0 + 16]; b1.y = W[(k + 1) * OUTD + col0 + 16];
        b2.x = W[k * OUTD + col0 + 32]; b2.y = W[(k + 1) * OUTD + col0 + 32];
        b3.x = W[k * OUTD + col0 + 48]; b3.y = W[(k + 1) * OUTD + col0 + 48];

        acc0 = __builtin_amdgcn_wmma_f32_16x16x4_f32(false, a, false, b0,
                                                     (short)0, acc0, false, false);
        acc1 = __builtin_amdgcn_wmma_f32_16x16x4_f32(false, a, false, b1,
                                                     (short)0, acc1, false, false);
        acc2 = __builtin_amdgcn_wmma_f32_16x16x4_f32(false, a, false, b2,
                                                     (short)0, acc2, false, false);
        acc3 = __builtin_amdgcn_wmma_f32_16x16x4_f32(false, a, false, b3,
                                                     (short)0, acc3, false, false);
    }

    // D layout: VGPR i -> M = i (lanes 0-15) / 8+i (lanes 16-31), N = lane&15
    const int mhalf = (lane >> 4) * 8;
    #pragma unroll
    for (int i = 0; i < 8; ++i) {
        const size_t row = (size_t)(mt * 16 + mhalf + i) * OUTD;
        h[row + col0 +  0] = acc0[i];
        h[row + col0 + 16] = acc1[i];
        h[row + col0 + 32] = acc2[i];
        h[row + col0 + 48] = acc3[i];
    }
}

// ---------------------------------------------------------------------------
// Kernel 3: per-node attention logit halves: a_src/a_dst[rn,h] = <h[rn,h,:], att>
// ---------------------------------------------------------------------------
__global__ void gat_attn_dots(const float* __restrict__ h,
                              const float* __restrict__ att_src,
                              const float* __restrict__ att_dst,
                              float* __restrict__ a_src,
                              float* __restrict__ a_dst) {
    int tid = blockIdx.x * blockDim.x + threadIdx.x;   // rn*H + hh
    if (tid >= NROWS * HHEADS) return;
    const int hh = tid & (HHEADS - 1);
    const int rn = tid >> 2;
    const float* hp = h + (size_t)rn * OUTD + hh * CDIM;
    const float* as = att_src + hh * CDIM;
    const float* ad = att_dst + hh * CDIM;
    float ss = 0.f, sd = 0.f;
    #pragma unroll 8
    for (int c = 0; c < CDIM; ++c) {
        const float v = hp[c];
        ss += v * as[c];
        sd += v * ad[c];
    }
    a_src[tid] = ss;
    a_dst[tid] = sd;
}

// ---------------------------------------------------------------------------
// Kernel 4: per (r,edge,head): ex = exp(leaky_relu(a_src[s]+a_dst[d])),
//           self-loop-masked originals -> 0; den[r,d,h] += ex
// ---------------------------------------------------------------------------
__global__ void gat_edge_exp(const int* __restrict__ ei,
                             const float* __restrict__ a_src,
                             const float* __restrict__ a_dst,
                             float* __restrict__ exb,
                             float* __restrict__ den) {
    int tid = blockIdx.x * blockDim.x + threadIdx.x;   // (r*E'+e)*H + hh
    if (tid >= RREP * EPRIME * HHEADS) return;
    const int hh = tid & (HHEADS - 1);
    const int re = tid >> 2;
    const int e  = re % EPRIME;
    const int r  = re / EPRIME;
    int s, d; bool selfm;
    if (e < NEDGE) { s = ei[e]; d = ei[NEDGE + e]; selfm = (s == d); }
    else           { s = e - NEDGE; d = s; selfm = false; }
    float ex;
    if (selfm) {
        ex = 0.0f;                                  // exp(NEG_INF)
    } else {
        float v = a_src[(r * NNODE + s) * HHEADS + hh]
                + a_dst[(r * NNODE + d) * HHEADS + hh];
        v = v > 0.0f ? v : NEG_SLOPE * v;           // leaky_relu
        ex = __expf(v);
    }
    exb[tid] = ex;
    atomicAdd(&den[(r * NNODE + d) * HHEADS + hh], ex);
}

// ---------------------------------------------------------------------------
// Kernel 5: aggregation. One wave per (r,edge): out[r,d,f] += (1/H)*alpha*h[s]
//   lane covers 8 of the 256 (h,f) message values -> global_atomic_add_f32.
// ---------------------------------------------------------------------------
__global__ void gat_aggregate(const int* __restrict__ ei,
                              const float* __restrict__ h,
                              const float* __restrict__ exb,
                              const float* __restrict__ den,
                              float* __restrict__ out) {
    const int wid  = blockIdx.x * (blockDim.x >> 5) + (threadIdx.x >> 5);
    if (wid >= RREP * EPRIME) return;
    const int lane = threadIdx.x & 31;
    const int e = wid % EPRIME;
    const int r = wid / EPRIME;
    int s, d;
    if (e < NEDGE) { s = ei[e]; d = ei[NEDGE + e]; }
    else           { s = e - NEDGE; d = s; }

    float alpha[HHEADS];
    #pragma unroll
    for (int hh = 0; hh < HHEADS; ++hh) {
        const float ex = exb[(r * EPRIME + e) * HHEADS + hh];
        const float dn = den[(r * NNODE + d) * HHEADS + hh];
        alpha[hh] = (1.0f / HHEADS) * ex / (dn + 1e-16f);
    }

    const float* hs = h + (size_t)(r * NNODE + s) * OUTD;
    float* op = out + (size_t)(r * NNODE + d) * FDIM;
    #pragma unroll
    for (int t = 0; t < 8; ++t) {
        const int idx = lane + t * 32;   // 0..255 over (h, f)
        const int hh  = idx >> 6;
        const int f   = idx & 63;
        atomicAdd(&op[f], alpha[hh] * hs[idx]);
    }
}

// ---------------------------------------------------------------------------
extern "C" void kernel_launch(void* const* d_in, const int* in_sizes, int n_in,
                              void* d_out, int out_size, void* d_ws, size_t ws_size,
                              hipStream_t stream) {
    const float* x       = (const float*)d_in[0];   // [4,12,1000,64]
    const int*   ei      = (const int*)d_in[1];     // [2,12000]
    const float* W       = (const float*)d_in[2];   // [64,256]
    const float* att_src = (const float*)d_in[3];   // [4,64]
    const float* att_dst = (const float*)d_in[4];   // [4,64]
    const float* bias    = (const float*)d_in[5];   // [64]
    float* out = (float*)d_out;

    // workspace layout (floats)
    float* ws    = (float*)d_ws;
    float* hbuf  = ws;                                   // 48000*256 = 12,288,000
    float* a_src = hbuf  + (size_t)NROWS * OUTD;         // 192,000
    float* a_dst = a_src + (size_t)NROWS * HHEADS;       // 192,000
    float* exb   = a_dst + (size_t)NROWS * HHEADS;       // 48*13000*4 = 2,496,000
    float* den   = exb   + (size_t)RREP * EPRIME * HHEADS; // 192,000

    // 1) init out = bias, den = 0
    {
        const int n = NROWS * FDIM;
        gat_init<<<(n + 255) / 256, 256, 0, stream>>>(out, den, bias);
    }
    // 2) projection via WMMA f32 16x16x4
    {
        dim3 grid(NROWS / 16);   // 3000 M-tiles
        dim3 block(128);         // 4 waves, each owns a 64-wide N strip
        gat_proj_wmma<<<grid, block, 0, stream>>>(x, W, hbuf);
    }
    // 3) attention dot products
    {
        const int n = NROWS * HHEADS;
        gat_attn_dots<<<(n + 255) / 256, 256, 0, stream>>>(hbuf, att_src, att_dst,
                                                           a_src, a_dst);
    }
    // 4) edge logits -> exp, denominator atomics
    {
        const int n = RREP * EPRIME * HHEADS;
        gat_edge_exp<<<(n + 255) / 256, 256, 0, stream>>>(ei, a_src, a_dst, exb, den);
    }
    // 5) weighted scatter-add aggregation (one wave per (r,edge))
    {
        const int waves = RREP * EPRIME;                 // 624,000
        const int wpb   = 256 / 32;                      // 8 waves per block
        gat_aggregate<<<(waves + wpb - 1) / wpb, 256, 0, stream>>>(ei, hbuf, exb,
                                                                   den, out);
    }
}